// GATWithJK_27212912787814
// MI455X (gfx1250) — compile-verified
//
#include <hip/hip_runtime.h>

#define NN   50000
#define EE   800000
#define ETOT (EE + NN)
#define GGR  512
#define DD0  18
#define KPAD 32
#define DHH  256
#define HHH  4
#define CCH  64
#define FCIN 768
#define NEGS 0.2f

typedef float v2f __attribute__((ext_vector_type(2)));
typedef float v8f __attribute__((ext_vector_type(8)));

// ---------- helpers ----------
__device__ __forceinline__ unsigned enc_f32(float f) {
    unsigned u = __float_as_uint(f);
    return (u & 0x80000000u) ? ~u : (u | 0x80000000u);
}
__device__ __forceinline__ float dec_f32(unsigned e) {
    unsigned u = (e & 0x80000000u) ? (e & 0x7fffffffu) : ~e;
    return __uint_as_float(u);
}

// ---------- input staging ----------
__global__ void build_x(const int* __restrict__ ids, const float* __restrict__ feats,
                        const float* __restrict__ emb, float* __restrict__ xp) {
    int t = blockIdx.x * blockDim.x + threadIdx.x;
    if (t >= NN * KPAD) return;
    int n = t >> 5, c = t & 31;
    float v = 0.f;
    if (c < 8)        v = emb[ids[n] * 8 + c];
    else if (c < DD0) v = feats[n * (DD0 - 8) + (c - 8)];
    xp[t] = v;
}

__global__ void pad_w0(const float* __restrict__ w, float* __restrict__ wp) {
    int t = blockIdx.x * blockDim.x + threadIdx.x;
    if (t >= KPAD * DHH) return;
    int k = t >> 8, c = t & 255;
    wp[t] = (k < DD0) ? w[k * DHH + c] : 0.f;
}

// ---------- fp32 WMMA GEMM: C[M x Ncols] = A[M x K] @ B[K x Ncols] (+bias, relu) ----------
// one wave computes a 16x64 tile via 4 accumulators; M%16==0, Ncols%64==0, K%4==0
__global__ void gemm_wmma(const float* __restrict__ A, int lda,
                          const float* __restrict__ B, int ldb,
                          float* __restrict__ Co, int ldc,
                          int M, int Ncols, int K,
                          const float* __restrict__ bias, int do_relu) {
    int lane = threadIdx.x & 31;
    int wave = (blockIdx.x * blockDim.x + threadIdx.x) >> 5;
    int colStrips = Ncols >> 6;
    int total = (M >> 4) * colStrips;
    if (wave >= total) return;               // uniform per wave -> EXEC stays full
    int rt = wave / colStrips, cs = wave - rt * colStrips;
    int row0 = rt << 4, col0 = cs << 6;
    int half = lane >> 4, l16 = lane & 15;

    v8f acc[4] = {v8f{0}, v8f{0}, v8f{0}, v8f{0}};
    // A 16x4 frag: lanes0-15 hold k,k+1; lanes16-31 hold k+2,k+3 (row = l16)
    const float* ap = A + (size_t)(row0 + l16) * lda + half * 2;
    // B 4x16 frag: vgpr0 = row(k+2*half), vgpr1 = next row; col = col0+16t+l16
    const float* bp = B + (size_t)(half * 2) * ldb + col0 + l16;
    const size_t bstep = (size_t)4 * ldb;

    for (int k0 = 0; k0 < K; k0 += 4) {
        // batch ALL fragment loads first -> single clause + one wait, then 4 WMMAs
        v2f a;  a[0]  = ap[0];  a[1]  = ap[1];
        v2f b0; b0[0] = bp[0];  b0[1] = bp[ldb];
        v2f b1; b1[0] = bp[16]; b1[1] = bp[16 + ldb];
        v2f b2; b2[0] = bp[32]; b2[1] = bp[32 + ldb];
        v2f b3; b3[0] = bp[48]; b3[1] = bp[48 + ldb];
        acc[0] = __builtin_amdgcn_wmma_f32_16x16x4_f32(false, a, false, b0, (short)0, acc[0], false, false);
        acc[1] = __builtin_amdgcn_wmma_f32_16x16x4_f32(false, a, false, b1, (short)0, acc[1], false, false);
        acc[2] = __builtin_amdgcn_wmma_f32_16x16x4_f32(false, a, false, b2, (short)0, acc[2], false, false);
        acc[3] = __builtin_amdgcn_wmma_f32_16x16x4_f32(false, a, false, b3, (short)0, acc[3], false, false);
        ap += 4;
        bp += bstep;
    }
#pragma unroll
    for (int t = 0; t < 4; ++t) {
        int col = col0 + t * 16 + l16;
        float bv = bias ? bias[col] : 0.f;
#pragma unroll
        for (int i = 0; i < 8; ++i) {
            int row = row0 + i + half * 8;   // C/D layout: vgpr i -> M=i (+8 for hi half)
            float v = acc[t][i] + bv;
            if (do_relu) v = v > 0.f ? v : 0.f;
            Co[(size_t)row * ldc + col] = v;
        }
    }
}

// ---------- attention scores per (node, head) ----------
__global__ void attn_scores(const float* __restrict__ xl, const float* __restrict__ a_src,
                            const float* __restrict__ a_dst,
                            float* __restrict__ as_, float* __restrict__ ad_) {
    int t = blockIdx.x * blockDim.x + threadIdx.x;
    if (t >= NN * HHH) return;
    int n = t >> 2, h = t & 3;
    const float* x = xl + (size_t)n * DHH + h * CCH;
    const float* s = a_src + h * CCH;
    const float* d = a_dst + h * CCH;
    float s1 = 0.f, s2 = 0.f;
    for (int c = 0; c < CCH; ++c) { float v = x[c]; s1 += v * s[c]; s2 += v * d[c]; }
    as_[t] = s1; ad_[t] = s2;
}

// ---------- edge passes ----------
__global__ void edge_max(const int* __restrict__ ei, const float* __restrict__ as_,
                         const float* __restrict__ ad_, float* __restrict__ lg,
                         unsigned* __restrict__ mx) {
    int t = blockIdx.x * blockDim.x + threadIdx.x;
    if (t >= ETOT * HHH) return;
    int e = t >> 2, h = t & 3;
    int s, d;
    if (e < EE) { s = ei[e]; d = ei[EE + e]; } else { s = d = e - EE; }
    float v = as_[s * HHH + h] + ad_[d * HHH + h];
    v = v > 0.f ? v : NEGS * v;              // leaky_relu
    lg[t] = v;
    atomicMax(mx + d * HHH + h, enc_f32(v)); // native u32 atomic max (monotonic encoding)
}

__global__ void edge_exp(const int* __restrict__ ei, float* __restrict__ lg,
                         const unsigned* __restrict__ mx, float* __restrict__ ss) {
    int t = blockIdx.x * blockDim.x + threadIdx.x;
    if (t >= ETOT * HHH) return;
    int e = t >> 2, h = t & 3;
    int d = (e < EE) ? ei[EE + e] : (e - EE);
    float m = dec_f32(mx[d * HHH + h]);
    float ex = __expf(lg[t] - m);
    lg[t] = ex;
    unsafeAtomicAdd(ss + d * HHH + h, ex);   // global_atomic_add_f32
}

// one wave32 per edge: 8 channels per lane, float4 gathers, f32 atomic scatter
__global__ void edge_scatter(const int* __restrict__ ei, const float* __restrict__ xl,
                             const float* __restrict__ num, const float* __restrict__ ss,
                             float* __restrict__ xs, int loff) {
    int wid  = (blockIdx.x * blockDim.x + threadIdx.x) >> 5;
    int lane = threadIdx.x & 31;
    if (wid >= ETOT) return;                 // uniform per wave
    int s, d;
    if (wid < EE) { s = ei[wid]; d = ei[EE + wid]; } else { s = d = wid - EE; }
    int h = lane >> 3;                       // 8 cols/lane -> single head per lane
    float alpha = num[(size_t)wid * HHH + h] / (ss[(size_t)d * HHH + h] + 1e-16f);
    const float4* xp = (const float4*)(xl + (size_t)s * DHH + lane * 8);
    float4 v0 = xp[0], v1 = xp[1];
    float* op = xs + (size_t)d * FCIN + loff + lane * 8;
    unsafeAtomicAdd(op + 0, v0.x * alpha);
    unsafeAtomicAdd(op + 1, v0.y * alpha);
    unsafeAtomicAdd(op + 2, v0.z * alpha);
    unsafeAtomicAdd(op + 3, v0.w * alpha);
    unsafeAtomicAdd(op + 4, v1.x * alpha);
    unsafeAtomicAdd(op + 5, v1.y * alpha);
    unsafeAtomicAdd(op + 6, v1.z * alpha);
    unsafeAtomicAdd(op + 7, v1.w * alpha);
}

// ---------- per-layer epilogues on XS strided region ----------
__global__ void zero_region(float* __restrict__ xs, int loff) {
    int n = blockIdx.y, c = threadIdx.x;     // blockDim.x == 256
    xs[(size_t)n * FCIN + loff + c] = 0.f;
}

__global__ void bias_relu_inplace(float* __restrict__ xs, const float* __restrict__ b, int loff) {
    int n = blockIdx.y, c = threadIdx.x;
    size_t idx = (size_t)n * FCIN + loff + c;
    float v = xs[idx] + b[c];
    xs[idx] = v > 0.f ? v : 0.f;
}

// ---------- pooling ----------
__global__ void pool_cnt(const int* __restrict__ batch, float* __restrict__ cnt) {
    int t = blockIdx.x * blockDim.x + threadIdx.x;
    if (t >= NN) return;
    unsafeAtomicAdd(cnt + batch[t], 1.0f);
}

__global__ void pool_sum(const int* __restrict__ batch, const float* __restrict__ xs,
                         float* __restrict__ sums) {
    int n = blockIdx.y;
    int c = blockIdx.x * blockDim.x + threadIdx.x;   // grid.x=3, blockDim=256 -> 768 exact
    unsafeAtomicAdd(sums + (size_t)batch[n] * FCIN + c, xs[(size_t)n * FCIN + c]);
}

__global__ void pool_div(const float* __restrict__ sums, const float* __restrict__ cnt,
                         float* __restrict__ z) {
    int t = blockIdx.x * blockDim.x + threadIdx.x;
    if (t >= GGR * FCIN) return;
    z[t] = sums[t] / fmaxf(cnt[t / FCIN], 1.0f);
}

// ---------- final 768 -> 2 ----------
__global__ void fc_last(const float* __restrict__ z, const float* __restrict__ W,
                        const float* __restrict__ b, float* __restrict__ out) {
    int t = blockIdx.x * blockDim.x + threadIdx.x;
    if (t >= GGR * 2) return;
    int g = t >> 1, o = t & 1;
    const float* zp = z + (size_t)g * FCIN;
    float acc = b[o];
    for (int k = 0; k < FCIN; ++k) acc += zp[k] * W[k * 2 + o];
    out[t] = acc;
}

extern "C" void kernel_launch(void* const* d_in, const int* in_sizes, int n_in,
                              void* d_out, int out_size, void* d_ws, size_t ws_size,
                              hipStream_t stream) {
    const int*   x_ids   = (const int*)d_in[0];
    const float* x_feats = (const float*)d_in[1];
    const int*   ei      = (const int*)d_in[2];
    const int*   batch   = (const int*)d_in[3];

    const float *emb, *Wc[3], *avsrc[3], *avdst[3], *bc[3], *fW[3], *fb[3];
    if (n_in >= 23 && in_sizes[4] == 2048 * 8) {
        // insertion order: emb, convs[(W,a_src,a_dst,b)], fc[(W,b)]
        emb = (const float*)d_in[4];
        int p = 5;
        for (int l = 0; l < 3; ++l) {
            Wc[l] = (const float*)d_in[p]; avsrc[l] = (const float*)d_in[p + 1];
            avdst[l] = (const float*)d_in[p + 2]; bc[l] = (const float*)d_in[p + 3]; p += 4;
        }
        for (int i = 0; i < 3; ++i) { fW[i] = (const float*)d_in[p]; fb[i] = (const float*)d_in[p + 1]; p += 2; }
    } else {
        // jax pytree (alphabetical) order: convs[(W,a_dst,a_src,b)], emb, fc[(W,b)]
        int p = 4;
        for (int l = 0; l < 3; ++l) {
            Wc[l] = (const float*)d_in[p]; avdst[l] = (const float*)d_in[p + 1];
            avsrc[l] = (const float*)d_in[p + 2]; bc[l] = (const float*)d_in[p + 3]; p += 4;
        }
        emb = (const float*)d_in[p]; ++p;
        for (int i = 0; i < 3; ++i) { fW[i] = (const float*)d_in[p]; fb[i] = (const float*)d_in[p + 1]; p += 2; }
    }

    // workspace layout (floats)
    float* w = (float*)d_ws;
    size_t off = 0;
    float*    Xpad  = w + off; off += (size_t)NN * KPAD;
    float*    W0pad = w + off; off += (size_t)KPAD * DHH;
    float*    XL    = w + off; off += (size_t)NN * DHH;
    float*    XS    = w + off; off += (size_t)NN * FCIN;
    float*    AS    = w + off; off += (size_t)NN * HHH;
    float*    AD    = w + off; off += (size_t)NN * HHH;
    float*    LG    = w + off; off += (size_t)ETOT * HHH;
    unsigned* MAXE  = (unsigned*)(w + off); off += (size_t)NN * HHH;
    float*    SSUM  = w + off; off += (size_t)NN * HHH;
    float*    SUMS  = w + off; off += (size_t)GGR * FCIN;
    float*    CNT   = w + off; off += (size_t)GGR;
    float*    Z0    = w + off; off += (size_t)GGR * FCIN;
    float*    Z1    = w + off; off += (size_t)GGR * FCIN;
    (void)ws_size; (void)out_size;

    build_x<<<(NN * KPAD + 255) / 256, 256, 0, stream>>>(x_ids, x_feats, emb, Xpad);
    pad_w0<<<(KPAD * DHH + 255) / 256, 256, 0, stream>>>(Wc[0], W0pad);

    for (int l = 0; l < 3; ++l) {
        const float* A = (l == 0) ? Xpad : (XS + (size_t)(l - 1) * DHH);
        int lda = (l == 0) ? KPAD : FCIN;
        const float* B = (l == 0) ? W0pad : Wc[l];
        int K = (l == 0) ? KPAD : DHH;
        int waves = (NN / 16) * (DHH / 64);          // 12500 waves
        gemm_wmma<<<(waves * 32 + 255) / 256, 256, 0, stream>>>(
            A, lda, B, DHH, XL, DHH, NN, DHH, K, nullptr, 0);

        attn_scores<<<(NN * HHH + 255) / 256, 256, 0, stream>>>(XL, avsrc[l], avdst[l], AS, AD);

        (void)hipMemsetAsync(MAXE, 0, (size_t)NN * HHH * 4, stream);   // enc sentinel == 0
        (void)hipMemsetAsync(SSUM, 0, (size_t)NN * HHH * 4, stream);
        zero_region<<<dim3(1, NN), 256, 0, stream>>>(XS, l * DHH);

        edge_max<<<(ETOT * HHH + 255) / 256, 256, 0, stream>>>(ei, AS, AD, LG, MAXE);
        edge_exp<<<(ETOT * HHH + 255) / 256, 256, 0, stream>>>(ei, LG, MAXE, SSUM);
        edge_scatter<<<(ETOT + 7) / 8, 256, 0, stream>>>(ei, XL, LG, SSUM, XS, l * DHH);

        bias_relu_inplace<<<dim3(1, NN), 256, 0, stream>>>(XS, bc[l], l * DHH);
    }

    (void)hipMemsetAsync(SUMS, 0, (size_t)GGR * FCIN * 4, stream);
    (void)hipMemsetAsync(CNT, 0, (size_t)GGR * 4, stream);
    pool_cnt<<<(NN + 255) / 256, 256, 0, stream>>>(batch, CNT);
    pool_sum<<<dim3(3, NN), 256, 0, stream>>>(batch, XS, SUMS);
    pool_div<<<(GGR * FCIN + 255) / 256, 256, 0, stream>>>(SUMS, CNT, Z0);

    int fwaves = (GGR / 16) * (FCIN / 64);           // 384 waves
    gemm_wmma<<<(fwaves * 32 + 255) / 256, 256, 0, stream>>>(
        Z0, FCIN, fW[0], FCIN, Z1, FCIN, GGR, FCIN, FCIN, fb[0], 1);
    gemm_wmma<<<(fwaves * 32 + 255) / 256, 256, 0, stream>>>(
        Z1, FCIN, fW[1], FCIN, Z0, FCIN, GGR, FCIN, FCIN, fb[1], 1);
    fc_last<<<(GGR * 2 + 255) / 256, 256, 0, stream>>>(Z0, fW[2], fb[2], (float*)d_out);
}